// SemanticDynamicBrainGraphMEG_80075370266892
// MI455X (gfx1250) — compile-verified
//
#include <hip/hip_runtime.h>
#include <hip/hip_bf16.h>
#include <math.h>

// Problem constants (from reference)
#define BB   256   // batch
#define CC   271   // channels (nodes)
#define TT   201   // time
#define DGN  256   // graph dim
#define KTOP 12
#define KSEM 2048
#define NSEM 4336  // C*16

typedef __attribute__((ext_vector_type(2))) float v2f;
typedef __attribute__((ext_vector_type(8))) float v8f;

// ---- single WMMA step: D = A(16x4 f32) * B(4x16 f32) + C (16x16 f32) ----
__device__ __forceinline__ v8f wmma4(v2f a, v2f b, v8f c) {
    // 8 args: (neg_a, A, neg_b, B, c_mod, C, reuse_a, reuse_b)
    return __builtin_amdgcn_wmma_f32_16x16x4_f32(false, a, false, b, (short)0, c,
                                                 false, false);
}

// =====================================================================
// Kernel 1: w[B][4336] = z[B][2048] @ W_sem[4336][2048]^T   (fp32 WMMA)
// grid = (NSEM/16, BB/16), block = 32 (one wave per 16x16 tile)
// =====================================================================
__global__ __launch_bounds__(32)
void sem_proj_kernel(const float* __restrict__ z, const float* __restrict__ Wsem,
                     float* __restrict__ w) {
    const int lane = threadIdx.x;
    const int col  = lane & 15;
    const int hi   = lane >> 4;
    const int n0   = blockIdx.x * 16;
    const int m0   = blockIdx.y * 16;

    const float* Arow = z    + (size_t)(m0 + col) * KSEM;  // A row (row idx == lane&15)
    const float* Brow = Wsem + (size_t)(n0 + col) * KSEM;  // B[k][n] = Wsem[n][k]

    v8f acc = {};
    for (int k = 0; k < KSEM; k += 4) {
        __builtin_prefetch(Brow + k + 128, 0, 0);
        v2f a = *(const v2f*)(Arow + k + 2 * hi);
        v2f b = *(const v2f*)(Brow + k + 2 * hi);
        acc = wmma4(a, b, acc);
    }
    for (int j = 0; j < 8; ++j) {
        int r = m0 + j + 8 * hi;
        w[(size_t)r * NSEM + n0 + col] = acc[j];
    }
}

// =====================================================================
// Kernel 2: S[b][i][j] = (w_b[i,:16] . w_b[j,:16]) * 0.25   (fp32 WMMA)
// grid = (ceil(C/16), ceil(C/16), B), block = 32
// =====================================================================
__global__ __launch_bounds__(32)
void gram_kernel(const float* __restrict__ w, float* __restrict__ S) {
    const int lane = threadIdx.x;
    const int col  = lane & 15;
    const int hi   = lane >> 4;
    const int j0   = blockIdx.x * 16;
    const int i0   = blockIdx.y * 16;
    const int b    = blockIdx.z;

    const float* wb = w + (size_t)b * NSEM;
    const int  ra   = i0 + col;
    const int  rb   = j0 + col;
    const bool aok  = ra < CC;
    const bool bok  = rb < CC;
    const float* Arow = wb + (size_t)(aok ? ra : 0) * 16;
    const float* Brow = wb + (size_t)(bok ? rb : 0) * 16;

    v8f acc = {};
    for (int k = 0; k < 16; k += 4) {
        v2f a; a.x = aok ? Arow[k + 2 * hi] : 0.f; a.y = aok ? Arow[k + 2 * hi + 1] : 0.f;
        v2f b2; b2.x = bok ? Brow[k + 2 * hi] : 0.f; b2.y = bok ? Brow[k + 2 * hi + 1] : 0.f;
        acc = wmma4(a, b2, acc);
    }
    for (int j = 0; j < 8; ++j) {
        int r = i0 + j + 8 * hi;
        int c = j0 + col;
        if (r < CC && c < CC)
            S[((size_t)b * CC + r) * CC + c] = acc[j] * 0.25f;
    }
}

// =====================================================================
// Kernel 3: per-row softmax + top-12 scatter.  One wave per (b, row).
// grid = (C, B), block = 32
// =====================================================================
__device__ __forceinline__ float wave_max(float v) {
    for (int off = 16; off > 0; off >>= 1)
        v = fmaxf(v, __shfl_xor(v, off, 32));
    return v;
}
__device__ __forceinline__ float wave_sum(float v) {
    for (int off = 16; off > 0; off >>= 1)
        v += __shfl_xor(v, off, 32);
    return v;
}

__global__ __launch_bounds__(32)
void softmax_topk_kernel(const float* __restrict__ S, float* __restrict__ Aout) {
    const int r    = blockIdx.x;
    const int b    = blockIdx.y;
    const int lane = threadIdx.x;
    const float* row  = S    + ((size_t)b * CC + r) * CC;
    float*       orow = Aout + ((size_t)b * CC + r) * CC;

    float v[9];
    float m = -__builtin_inff();
    #pragma unroll
    for (int i = 0; i < 9; ++i) {
        int idx = lane + 32 * i;
        v[i] = (idx < CC) ? row[idx] : -__builtin_inff();
        m = fmaxf(m, v[i]);
    }
    m = wave_max(m);

    float s = 0.f;
    #pragma unroll
    for (int i = 0; i < 9; ++i) {
        int idx = lane + 32 * i;
        v[i] = (idx < CC) ? __expf(v[i] - m) : 0.f;
        s += v[i];
    }
    s = wave_sum(s);
    const float inv = 1.f / s;
    #pragma unroll
    for (int i = 0; i < 9; ++i) v[i] *= inv;

    // zero the output row
    #pragma unroll
    for (int i = 0; i < 9; ++i) {
        int idx = lane + 32 * i;
        if (idx < CC) orow[idx] = 0.f;
    }

    // 12 rounds of wave-argmax extraction
    for (int t = 0; t < KTOP; ++t) {
        float lv = -1.f;
        int   li = CC;
        #pragma unroll
        for (int i = 0; i < 9; ++i) {
            int idx = lane + 32 * i;
            if (idx < CC && v[i] > lv) { lv = v[i]; li = idx; }
        }
        for (int off = 16; off > 0; off >>= 1) {
            float ov = __shfl_xor(lv, off, 32);
            int   oi = __shfl_xor(li, off, 32);
            if (ov > lv || (ov == lv && oi < li)) { lv = ov; li = oi; }
        }
        if (lane == (li & 31)) {
            orow[li] = lv;
            v[li >> 5] = -1.f;   // remove from further consideration
        }
    }
}

// =====================================================================
// Kernel 4: symmetrize A in place: A = (A + A^T) * 0.5
// =====================================================================
__global__ __launch_bounds__(256)
void symmetrize_kernel(float* __restrict__ A) {
    size_t tid   = (size_t)blockIdx.x * blockDim.x + threadIdx.x;
    size_t total = (size_t)BB * CC * CC;
    if (tid >= total) return;
    int j = (int)(tid % CC);
    int i = (int)((tid / CC) % CC);
    int b = (int)(tid / ((size_t)CC * CC));
    if (i > j) return;
    float* Ab = A + (size_t)b * CC * CC;
    float x = Ab[(size_t)i * CC + j];
    float y = Ab[(size_t)j * CC + i];
    float m = 0.5f * (x + y);
    Ab[(size_t)i * CC + j] = m;
    Ab[(size_t)j * CC + i] = m;
}

// =====================================================================
// Kernel 5: h[b][c][d] = x[b][c][:] . W_temp[d][:] + b_temp[d]  (fp32 WMMA)
// 16(c) x 64(d) per wave: A fragment reused across 4 d-subtiles.
// grid = (DG/64, ceil(C/16), B), block = 32
// =====================================================================
__global__ __launch_bounds__(32)
void temporal_kernel(const float* __restrict__ x, const float* __restrict__ Wt,
                     const float* __restrict__ bt, float* __restrict__ h) {
    const int lane = threadIdx.x;
    const int col  = lane & 15;
    const int hi   = lane >> 4;
    const int d0   = blockIdx.x * 64;
    const int c0   = blockIdx.y * 16;
    const int b    = blockIdx.z;

    const int  ca  = c0 + col;
    const bool aok = ca < CC;
    const float* Arow = x + ((size_t)b * CC + (aok ? ca : 0)) * TT;
    const float* Brow[4];
    #pragma unroll
    for (int q = 0; q < 4; ++q)
        Brow[q] = Wt + (size_t)(d0 + 16 * q + col) * TT;   // B[t][d] = Wt[d][t]

    v8f acc[4] = {};
    for (int k = 0; k < TT; k += 4) {      // last iter k=200 covers t=200..203 (guarded)
        int t0 = k + 2 * hi;
        v2f a;
        a.x = (aok && t0     < TT) ? Arow[t0]     : 0.f;
        a.y = (aok && t0 + 1 < TT) ? Arow[t0 + 1] : 0.f;
        #pragma unroll
        for (int q = 0; q < 4; ++q) {
            v2f bb;
            bb.x = (t0     < TT) ? Brow[q][t0]     : 0.f;
            bb.y = (t0 + 1 < TT) ? Brow[q][t0 + 1] : 0.f;
            acc[q] = wmma4(a, bb, acc[q]);
        }
    }
    #pragma unroll
    for (int q = 0; q < 4; ++q) {
        const float bias = bt[d0 + 16 * q + col];
        for (int j = 0; j < 8; ++j) {
            int r = c0 + j + 8 * hi;
            if (r < CC)
                h[((size_t)b * CC + r) * DGN + d0 + 16 * q + col] = acc[q][j] + bias;
        }
    }
}

// =====================================================================
// Kernel 6: h2[b][c][d] = sum_e A[b][c][e] * h[b][e][d]   (fp32 WMMA)
// 4 waves/block; block computes 16(c) x 64(d). The h K-panel (32e x 64d)
// is staged in LDS with GLOBAL_LOAD_ASYNC_TO_LDS_B128 (ASYNCcnt tracked),
// then all 4 waves consume it via DS reads into WMMA B fragments.
// grid = (DG/64, ceil(C/16), B), block = 128
// =====================================================================
#define KB 32   // e-chunk staged per iteration

__global__ __launch_bounds__(128)
void aggregate_kernel(const float* __restrict__ A, const float* __restrict__ h,
                      float* __restrict__ h2) {
    __shared__ float tileB[KB * 64];     // 8 KB: h[e0:e0+32][d0:d0+64]

    const int tid  = threadIdx.x;
    const int lane = tid & 31;
    const int wv   = tid >> 5;           // wave id 0..3 -> d subtile
    const int col  = lane & 15;
    const int hi   = lane >> 4;
    const int d0   = blockIdx.x * 64;
    const int c0   = blockIdx.y * 16;
    const int b    = blockIdx.z;
    const int n0   = 16 * wv;            // d offset within the 64-wide tile

    const int  ca  = c0 + col;
    const bool aok = ca < CC;
    const float* Arow = A + ((size_t)b * CC + (aok ? ca : 0)) * CC;
    const float* Hb   = h + (size_t)b * CC * DGN;

    v8f acc = {};
    for (int e0 = 0; e0 < CC; e0 += KB) {
        // ragged final chunk: zero-fill the LDS panel first (uniform branch)
        if (e0 + KB > CC) {
            #pragma unroll
            for (int i = 0; i < 16; ++i) tileB[tid * 16 + i] = 0.f;
            __syncthreads();
        }
        // async-stage h[e0:e0+KB][d0:d0+64] into LDS (4 x b128 per thread)
        #pragma unroll
        for (int i = 0; i < 4; ++i) {
            int q    = tid + 128 * i;      // 0..511 b128 chunks
            int row  = q >> 4;             // 0..31 (e-local)
            int colw = (q & 15) * 4;       // float col within 64-wide panel
            int e    = e0 + row;
            if (e < CC) {
                const float* gp = Hb + (size_t)e * DGN + d0 + colw;
                unsigned int lo = (unsigned int)(uintptr_t)&tileB[row * 64 + colw];
                asm volatile("global_load_async_to_lds_b128 %0, %1, off"
                             :: "v"(lo), "v"(gp) : "memory");
            }
        }
        asm volatile("s_wait_asynccnt 0x0" ::: "memory");
        __syncthreads();

        // consume: 8 WMMA steps over the staged panel
        #pragma unroll
        for (int kk = 0; kk < KB; kk += 4) {
            int el = kk + 2 * hi;
            int e  = e0 + el;
            v2f a, bb;
            a.x  = (aok && e     < CC) ? Arow[e]     : 0.f;
            a.y  = (aok && e + 1 < CC) ? Arow[e + 1] : 0.f;
            bb.x = tileB[el * 64 + n0 + col];
            bb.y = tileB[(el + 1) * 64 + n0 + col];
            acc = wmma4(a, bb, acc);
        }
        __syncthreads();   // protect LDS panel before next staging round
    }

    for (int j = 0; j < 8; ++j) {
        int r = c0 + j + 8 * hi;
        if (r < CC)
            h2[((size_t)b * CC + r) * DGN + d0 + n0 + col] = acc[j];
    }
}

// =====================================================================
// Kernel 7: attention pooling. One 256-thread block per batch.
// =====================================================================
__global__ __launch_bounds__(256)
void pool_kernel(const float* __restrict__ h2, const float* __restrict__ W1,
                 const float* __restrict__ b1, const float* __restrict__ W2,
                 const float* __restrict__ b2, float* __restrict__ zg) {
    const int b = blockIdx.x;
    const int t = threadIdx.x;
    const float* H = h2 + (size_t)b * CC * DGN;

    __shared__ float sc[CC];
    __shared__ float red[256];

    // per-channel attention score
    for (int c = t; c < CC; c += 256) {
        const float* hr = H + (size_t)c * DGN;
        float s = b2[0];
        for (int hh = 0; hh < DGN / 2; ++hh) {
            const float* w1r = W1 + (size_t)hh * DGN;
            float acc = b1[hh];
            for (int d = 0; d < DGN; ++d) acc = fmaf(hr[d], w1r[d], acc);
            s = fmaf(tanhf(acc), W2[hh], s);
        }
        sc[c] = s;
    }
    __syncthreads();

    // softmax over channels
    float m = -__builtin_inff();
    for (int c = t; c < CC; c += 256) m = fmaxf(m, sc[c]);
    red[t] = m;
    __syncthreads();
    for (int off = 128; off > 0; off >>= 1) {
        if (t < off) red[t] = fmaxf(red[t], red[t + off]);
        __syncthreads();
    }
    const float rmax = red[0];
    __syncthreads();

    float ssum = 0.f;
    for (int c = t; c < CC; c += 256) {
        float e = __expf(sc[c] - rmax);
        sc[c] = e;
        ssum += e;
    }
    red[t] = ssum;
    __syncthreads();
    for (int off = 128; off > 0; off >>= 1) {
        if (t < off) red[t] += red[t + off];
        __syncthreads();
    }
    const float inv = 1.f / red[0];
    __syncthreads();

    // z_graph[b][d], d = t (DGN == 256 == blockDim)
    float acc = 0.f;
    for (int c = 0; c < CC; ++c) acc = fmaf(sc[c], H[(size_t)c * DGN + t], acc);
    zg[(size_t)b * DGN + t] = acc * inv;
}

// =====================================================================
// Host launcher
// =====================================================================
extern "C" void kernel_launch(void* const* d_in, const int* in_sizes, int n_in,
                              void* d_out, int out_size, void* d_ws, size_t ws_size,
                              hipStream_t stream) {
    const float* x     = (const float*)d_in[0];   // [B, C, T]
    const float* z2048 = (const float*)d_in[1];   // [B, 2048]
    const float* Wsem  = (const float*)d_in[2];   // [4336, 2048]
    const float* Wt    = (const float*)d_in[3];   // [DG, T]
    const float* bt    = (const float*)d_in[4];   // [DG]
    const float* W1    = (const float*)d_in[5];   // [128, DG]
    const float* b1    = (const float*)d_in[6];   // [128]
    const float* W2    = (const float*)d_in[7];   // [1, 128]
    const float* b2    = (const float*)d_in[8];   // [1]

    float* out  = (float*)d_out;
    float* zg   = out;                    // [B, DG] = 65536 floats
    float* Aout = out + (size_t)BB * DGN; // [B, C, C]

    // workspace layout (256B aligned)
    auto align256 = [](size_t v) { return (v + 255) & ~(size_t)255; };
    char*  ws    = (char*)d_ws;
    size_t off_w = 0;
    size_t off_S = align256(off_w + (size_t)BB * NSEM * sizeof(float));
    size_t off_h = align256(off_S + (size_t)BB * CC * CC * sizeof(float));
    float* w  = (float*)(ws + off_w);
    float* S  = (float*)(ws + off_S);
    float* h  = (float*)(ws + off_h);
    float* h2 = S;   // S is dead after top-k; reuse for h2

    const int ctile = (CC + 15) / 16;  // 17

    sem_proj_kernel<<<dim3(NSEM / 16, BB / 16), 32, 0, stream>>>(z2048, Wsem, w);
    gram_kernel<<<dim3(ctile, ctile, BB), 32, 0, stream>>>(w, S);
    softmax_topk_kernel<<<dim3(CC, BB), 32, 0, stream>>>(S, Aout);
    {
        size_t total = (size_t)BB * CC * CC;
        int blocks = (int)((total + 255) / 256);
        symmetrize_kernel<<<blocks, 256, 0, stream>>>(Aout);
    }
    temporal_kernel<<<dim3(DGN / 64, ctile, BB), 32, 0, stream>>>(x, Wt, bt, h);
    aggregate_kernel<<<dim3(DGN / 64, ctile, BB), 128, 0, stream>>>(Aout, h, h2);
    pool_kernel<<<BB, 256, 0, stream>>>(h2, W1, b1, W2, b2, zg);
}